// SpectralConv_72919954751499
// MI455X (gfx1250) — compile-verified
//
#include <hip/hip_runtime.h>

typedef __attribute__((ext_vector_type(2))) float v2f;
typedef __attribute__((ext_vector_type(8))) float v8f;

#define TWO_PI 6.283185307179586f

// ---------------- WMMA helpers (V_WMMA_F32_16X16X4_F32) ----------------
// A 16x4 (MxK): lane&15 = M, K = v + 2*(lane>>4)  -> consecutive k => float2
// B 4x16 (KxN): lane&15 = N, K = v + 2*(lane>>4)
// C/D 16x16:    lane&15 = N, M = r + 8*(lane>>4)

__device__ __forceinline__ v8f wmma4(v2f a, v2f b, v8f c) {
  return __builtin_amdgcn_wmma_f32_16x16x4_f32(
      false, a, false, b, (short)0, c, false, false);
}

__device__ __forceinline__ v2f load_a(const float* __restrict__ A, int lda,
                                      int m0, int k0, int lane) {
  int m = m0 + (lane & 15);
  int k = k0 + 2 * (lane >> 4);
  const float* p = A + (size_t)m * lda + k;
  v2f r; r.x = p[0]; r.y = p[1];
  return r;
}

__device__ __forceinline__ v2f load_b(const float* __restrict__ B, int ldb,
                                      int k0, int n0, int lane) {
  int n = n0 + (lane & 15);
  int k = k0 + 2 * (lane >> 4);
  v2f r; r.x = B[(size_t)k * ldb + n]; r.y = B[(size_t)(k + 1) * ldb + n];
  return r;
}

__device__ __forceinline__ void store_c(float* __restrict__ C, int ldc,
                                        int m0, int n0, v8f acc, int lane) {
  int n = n0 + (lane & 15);
  int mb = m0 + 8 * (lane >> 4);
#pragma unroll
  for (int r = 0; r < 8; ++r) C[(size_t)(mb + r) * ldc + n] = acc[r];
}

// ---------------- twiddle tables ----------------
// E1 [64][256] : rows 0..31 = s*cos(2pi ky w/256), rows 32..63 = -s*sin, s=1/65536
// Fre/Fim [64][256] : kx rows (0..31, 224..255), e^{-i t}: cos, -sin
// Gre/Gim [256][64] : e^{+i t}: cos, +sin   (h major)
// Wm [256][64] : cols 0..31 = c*cos(2pi ky w/256), 32..63 = -c*sin ; c=1 (ky=0) else 2
__global__ void init_tables(float* __restrict__ E1, float* __restrict__ Fre,
                            float* __restrict__ Fim, float* __restrict__ Gre,
                            float* __restrict__ Gim, float* __restrict__ Wm) {
  int idx = blockIdx.x * blockDim.x + threadIdx.x;
  if (idx >= 16384) return;
  {
    int row = idx >> 8, w = idx & 255;
    int ky = row & 31;
    float th = TWO_PI * (float)((ky * w) & 255) / 256.0f;
    const float s = 1.0f / 65536.0f;
    E1[idx] = (row < 32) ? s * cosf(th) : -s * sinf(th);
  }
  {
    int r = idx >> 8, h = idx & 255;
    int kx = (r < 32) ? r : (192 + r);
    float th = TWO_PI * (float)((kx * h) & 255) / 256.0f;
    Fre[idx] = cosf(th);
    Fim[idx] = -sinf(th);
  }
  {
    int h = idx >> 6, r = idx & 63;
    int kx = (r < 32) ? r : (192 + r);
    float th = TWO_PI * (float)((kx * h) & 255) / 256.0f;
    Gre[idx] = cosf(th);
    Gim[idx] = sinf(th);
  }
  {
    int w = idx >> 6, k = idx & 63;
    int ky = k & 31;
    float c = (ky == 0) ? 1.0f : 2.0f;
    float th = TWO_PI * (float)((ky * w) & 255) / 256.0f;
    Wm[idx] = (k < 32) ? c * cosf(th) : -c * sinf(th);
  }
}

// ---------------- stage 1: pruned W-DFT (real GEMM) ----------------
// One wave computes the full 64x64 T block of one (b,h):
// T[bh][row(64: re0..31,im0..31)][c] = sum_w E1[row][w] * x[bh][w][c]
__global__ __launch_bounds__(256) void stage1(const float* __restrict__ x,
                                              const float* __restrict__ E1,
                                              float* __restrict__ T) {
  int lane = threadIdx.x & 31;
  int bh = blockIdx.x * 8 + (threadIdx.x >> 5);  // 4096 waves
  const float* Bmat = x + (size_t)bh * (256 * 64);
  v8f acc[4][4] = {};
  for (int k0 = 0; k0 < 256; k0 += 4) {
    v2f a[4], b[4];
#pragma unroll
    for (int t = 0; t < 4; ++t) a[t] = load_a(E1, 256, t * 16, k0, lane);
#pragma unroll
    for (int t = 0; t < 4; ++t) b[t] = load_b(Bmat, 64, k0, t * 16, lane);
#pragma unroll
    for (int i = 0; i < 4; ++i)
#pragma unroll
      for (int j = 0; j < 4; ++j) acc[i][j] = wmma4(a[i], b[j], acc[i][j]);
  }
  float* Tb = T + (size_t)bh * (64 * 64);
#pragma unroll
  for (int i = 0; i < 4; ++i)
#pragma unroll
    for (int j = 0; j < 4; ++j) store_c(Tb, 64, i * 16, j * 16, acc[i][j], lane);
}

// ---------------- stage 2: pruned H-DFT (complex GEMM, 3M) ----------------
// One wave computes a 64(r) x 16(c) strip of one (b,ky).
// Y[b][ky][part][r][c] ; per (b,ky): F(64x256) * T(256x64)
__global__ __launch_bounds__(256) void stage2(const float* __restrict__ T,
                                              const float* __restrict__ Fre,
                                              const float* __restrict__ Fim,
                                              float* __restrict__ Y) {
  int lane = threadIdx.x & 31;
  int wid = blockIdx.x * 8 + (threadIdx.x >> 5);  // B*32*4 = 2048 waves
  int tn = (wid & 3) * 16;
  int m = wid >> 2;  // b*32+ky
  int b = m >> 5, ky = m & 31;
  const float* Tre = T + (size_t)b * 256 * 64 * 64 + (size_t)ky * 64;
  const float* Tim = Tre + (size_t)32 * 64;
  v8f t1[4] = {}, t2[4] = {}, t3[4] = {};
  for (int k0 = 0; k0 < 256; k0 += 4) {
    v2f ar[4], ai[4], as[4];
#pragma unroll
    for (int t = 0; t < 4; ++t) {
      ar[t] = load_a(Fre, 256, t * 16, k0, lane);
      ai[t] = load_a(Fim, 256, t * 16, k0, lane);
      as[t] = ar[t] + ai[t];
    }
    v2f br = load_b(Tre, 4096, k0, tn, lane);  // h-stride = 64*64
    v2f bi = load_b(Tim, 4096, k0, tn, lane);
    v2f bs = br + bi;
#pragma unroll
    for (int t = 0; t < 4; ++t) {
      t1[t] = wmma4(ar[t], br, t1[t]);
      t2[t] = wmma4(ai[t], bi, t2[t]);
      t3[t] = wmma4(as[t], bs, t3[t]);
    }
  }
  float* Yb = Y + (size_t)m * (2 * 64 * 64);
#pragma unroll
  for (int t = 0; t < 4; ++t) {
    v8f cre = t1[t] - t2[t];
    v8f cim = t3[t] - t1[t] - t2[t];
    store_c(Yb, 64, t * 16, tn, cre, lane);
    store_c(Yb + 64 * 64, 64, t * 16, tn, cim, lane);
  }
}

// ---------------- stage 3: per-mode complex channel mix (3M) ----------------
// One wave computes the full 16(b) x 64(o) block of one mode (r,ky).
__global__ __launch_bounds__(256) void stage3(const float* __restrict__ Y,
                                              const float* __restrict__ w1re,
                                              const float* __restrict__ w1im,
                                              const float* __restrict__ w2re,
                                              const float* __restrict__ w2im,
                                              float* __restrict__ Yc) {
  int lane = threadIdx.x & 31;
  int mode = blockIdx.x * 8 + (threadIdx.x >> 5);  // 64*32 = 2048 waves
  int r = mode >> 5, ky = mode & 31;
  const float* wre;
  const float* wim;
  int rr;
  if (r < 32) { wre = w1re; wim = w1im; rr = r; }
  else        { wre = w2re; wim = w2im; rr = r - 32; }
  const float* Are = Y + ((size_t)ky * 128 + r) * 64;  // + b*262144 + i
  const float* Aim = Are + (size_t)64 * 64;
  size_t wbase = (size_t)rr * 32 + ky;  // + (i*64+o)*1024
  int mb = lane & 15;
  int nl = lane & 15;
  v8f t1[4] = {}, t2[4] = {}, t3[4] = {};
  for (int k0 = 0; k0 < 64; k0 += 4) {
    int kk = k0 + 2 * (lane >> 4);
    const float* pre = Are + (size_t)mb * 262144 + kk;
    const float* pim = Aim + (size_t)mb * 262144 + kk;
    v2f ar; ar.x = pre[0]; ar.y = pre[1];
    v2f ai; ai.x = pim[0]; ai.y = pim[1];
    v2f as = ar + ai;
#pragma unroll
    for (int t = 0; t < 4; ++t) {
      int nn = t * 16 + nl;
      size_t wb0 = wbase + ((size_t)kk * 64 + nn) * 1024;
      size_t wb1 = wbase + ((size_t)(kk + 1) * 64 + nn) * 1024;
      v2f br; br.x = wre[wb0]; br.y = wre[wb1];
      v2f bi; bi.x = wim[wb0]; bi.y = wim[wb1];
      v2f bs = br + bi;
      t1[t] = wmma4(ar, br, t1[t]);
      t2[t] = wmma4(ai, bi, t2[t]);
      t3[t] = wmma4(as, bs, t3[t]);
    }
  }
  int b0 = 8 * (lane >> 4);
#pragma unroll
  for (int t = 0; t < 4; ++t) {
    v8f cre = t1[t] - t2[t];
    v8f cim = t3[t] - t1[t] - t2[t];
    int nn = t * 16 + nl;
#pragma unroll
    for (int q = 0; q < 8; ++q) {
      int b = b0 + q;
      size_t addr = ((((size_t)b * 32 + ky) * 2) * 64 + r) * 64 + nn;
      Yc[addr] = cre[q];
      Yc[addr + 64 * 64] = cim[q];
    }
  }
}

// ---------------- stage 4: pruned H-inverse (complex GEMM, 3M) ----------------
// One wave computes a 64(h) x 16(o) strip of one (b,ky).
// per (b,ky): Z(256h x 64o) = G(256x64) * Yc(64x64)
__global__ __launch_bounds__(256) void stage4(const float* __restrict__ Yc,
                                              const float* __restrict__ Gre,
                                              const float* __restrict__ Gim,
                                              float* __restrict__ Z) {
  int lane = threadIdx.x & 31;
  int wid = blockIdx.x * 8 + (threadIdx.x >> 5);  // B*32*16 = 8192 waves
  int s = wid & 15;
  int m = wid >> 4;  // b*32+ky
  int b = m >> 5, ky = m & 31;
  int hs = (s >> 2) * 64;   // h strip base
  int tn = (s & 3) * 16;    // o tile base
  const float* Bre = Yc + (size_t)m * (2 * 64 * 64);
  const float* Bim = Bre + 64 * 64;
  v8f t1[4] = {}, t2[4] = {}, t3[4] = {};
  for (int k0 = 0; k0 < 64; k0 += 4) {
    v2f ar[4], ai[4], as[4];
#pragma unroll
    for (int t = 0; t < 4; ++t) {
      ar[t] = load_a(Gre, 64, hs + t * 16, k0, lane);
      ai[t] = load_a(Gim, 64, hs + t * 16, k0, lane);
      as[t] = ar[t] + ai[t];
    }
    v2f br = load_b(Bre, 64, k0, tn, lane);
    v2f bi = load_b(Bim, 64, k0, tn, lane);
    v2f bs = br + bi;
#pragma unroll
    for (int t = 0; t < 4; ++t) {
      t1[t] = wmma4(ar[t], br, t1[t]);
      t2[t] = wmma4(ai[t], bi, t2[t]);
      t3[t] = wmma4(as[t], bs, t3[t]);
    }
  }
  // Z[b][h][part][ky][o] ; per (b,h) block = 4096 floats
  int nn = tn + (lane & 15);
#pragma unroll
  for (int t = 0; t < 4; ++t) {
    v8f cre = t1[t] - t2[t];
    v8f cim = t3[t] - t1[t] - t2[t];
    int h0 = hs + t * 16 + 8 * (lane >> 4);
#pragma unroll
    for (int q = 0; q < 8; ++q) {
      int h = h0 + q;
      size_t addr = ((size_t)b * 256 + h) * 4096 + (size_t)ky * 64 + nn;
      Z[addr] = cre[q];
      Z[addr + 2048] = cim[q];
    }
  }
}

// ---------------- stage 5: pruned W-inverse (real GEMM) + bias ----------------
// One wave computes a 64(w) x 64(o) block of one (b,h); 4 strips/bh share a block.
__global__ __launch_bounds__(256) void stage5(const float* __restrict__ Z,
                                              const float* __restrict__ Wm,
                                              const float* __restrict__ bias,
                                              float* __restrict__ out) {
  int lane = threadIdx.x & 31;
  int wid = blockIdx.x * 8 + (threadIdx.x >> 5);  // B*H*4 = 16384 waves
  int strip = wid & 3;
  int bh = wid >> 2;
  int tm0 = strip * 64;
  const float* Bmat = Z + (size_t)bh * 4096;
  v8f acc[4][4] = {};
  for (int k0 = 0; k0 < 64; k0 += 4) {
    v2f a[4], b[4];
#pragma unroll
    for (int t = 0; t < 4; ++t) a[t] = load_a(Wm, 64, tm0 + t * 16, k0, lane);
#pragma unroll
    for (int t = 0; t < 4; ++t) b[t] = load_b(Bmat, 64, k0, t * 16, lane);
#pragma unroll
    for (int i = 0; i < 4; ++i)
#pragma unroll
      for (int j = 0; j < 4; ++j) acc[i][j] = wmma4(a[i], b[j], acc[i][j]);
  }
  float* O = out + (size_t)bh * (256 * 64);
  int nl = lane & 15;
#pragma unroll
  for (int j = 0; j < 4; ++j) {
    int nn = j * 16 + nl;
    float bv = bias[nn];
#pragma unroll
    for (int i = 0; i < 4; ++i) {
      int w0 = tm0 + i * 16 + 8 * (lane >> 4);
#pragma unroll
      for (int q = 0; q < 8; ++q) O[(size_t)(w0 + q) * 64 + nn] = acc[i][j][q] + bv;
    }
  }
}

extern "C" void kernel_launch(void* const* d_in, const int* in_sizes, int n_in,
                              void* d_out, int out_size, void* d_ws, size_t ws_size,
                              hipStream_t stream) {
  const float* x    = (const float*)d_in[0];
  const float* w1re = (const float*)d_in[1];
  const float* w1im = (const float*)d_in[2];
  const float* w2re = (const float*)d_in[3];
  const float* w2im = (const float*)d_in[4];
  const float* bias = (const float*)d_in[5];
  float* out = (float*)d_out;

  float* ws  = (float*)d_ws;
  float* E1  = ws;
  float* Fre = E1 + 16384;
  float* Fim = Fre + 16384;
  float* Gre = Fim + 16384;
  float* Gim = Gre + 16384;
  float* Wm  = Gim + 16384;
  float* T   = Wm + 16384;          // 16*256*64*64 = 16777216 floats
  float* Z   = T;                   // reuse (T dead after stage 2)
  float* Y   = T + 16777216;        // 8388608 floats
  float* Yc  = Y + 8388608;         // 8388608 floats

  init_tables<<<64, 256, 0, stream>>>(E1, Fre, Fim, Gre, Gim, Wm);
  stage1<<<512, 256, 0, stream>>>(x, E1, T);                  // 4096 waves
  stage2<<<256, 256, 0, stream>>>(T, Fre, Fim, Y);            // 2048 waves
  stage3<<<256, 256, 0, stream>>>(Y, w1re, w1im, w2re, w2im, Yc);  // 2048 waves
  stage4<<<1024, 256, 0, stream>>>(Yc, Gre, Gim, Z);          // 8192 waves
  stage5<<<2048, 256, 0, stream>>>(Z, Wm, bias, out);         // 16384 waves
}